// pyGraphSAGE_35957466202228
// MI455X (gfx1250) — compile-verified
//
#include <hip/hip_runtime.h>

typedef __attribute__((ext_vector_type(2))) float v2f;
typedef __attribute__((ext_vector_type(8))) float v8f;

#define IN_DIM 512
#define HID    32
#define OUT_DIM 16

// LDS layout (floats):
//   sW1 : [0, 16384)            512x32 weight matrix (64 KB)
//   sH  : [16384, 20480)        8 waves x (16x32) fc1 outputs (16 KB)
//   sW2 : [20480, 20992)        32x16
//   sb1 : [20992, 21024)
//   sb2 : [21024, 21040)
#define SMEM_FLOATS 21040
#define SMEM_BYTES  (SMEM_FLOATS * 4)

// Fused fc1 (WMMA f32) + ReLU + fc2 (VALU) -> x[N,16]
// 256 threads = 8 waves; each wave computes a 16-row node tile via
// V_WMMA_F32_16X16X4_F32 over K=512 (128 steps, 2 N-tiles for HID=32).
__global__ __launch_bounds__(256)
void mlp_fused_kernel(const float* __restrict__ feat,
                      const float* __restrict__ W1,
                      const float* __restrict__ b1,
                      const float* __restrict__ W2,
                      const float* __restrict__ b2,
                      float* __restrict__ x, int n)
{
    extern __shared__ float smem[];
    float* sW1 = smem;
    float* sH  = smem + 16384;
    float* sW2 = smem + 20480;
    float* sb1 = smem + 20992;
    float* sb2 = smem + 21024;

    // Cooperative stage of W1/W2/biases into LDS (read once per block).
    for (int i = threadIdx.x; i < IN_DIM * HID; i += 256) sW1[i] = W1[i];
    for (int i = threadIdx.x; i < HID * OUT_DIM; i += 256) sW2[i] = W2[i];
    if (threadIdx.x < HID)     sb1[threadIdx.x] = b1[threadIdx.x];
    if (threadIdx.x < OUT_DIM) sb2[threadIdx.x] = b2[threadIdx.x];
    __syncthreads();

    const int wave = threadIdx.x >> 5;
    const int lane = threadIdx.x & 31;
    const int row0 = blockIdx.x * 128 + wave * 16;

    if (row0 < n) {
        // A-matrix layout for V_WMMA_F32_16X16X4_F32 (16x4 fp32):
        //   lanes 0-15:  M=lane,    VGPR0=K0, VGPR1=K1
        //   lanes 16-31: M=lane-16, VGPR0=K2, VGPR1=K3
        const int m     = lane & 15;
        const int khalf = (lane >> 4) << 1;      // 0 or 2
        const int n0    = lane & 15;             // B/C/D column
        const float* arow = feat + (size_t)(row0 + m) * IN_DIM + khalf;

        v8f acc0 = {};   // HID columns 0..15
        v8f acc1 = {};   // HID columns 16..31

        #pragma unroll 4
        for (int k = 0; k < IN_DIM; k += 4) {
            // Stream-once 205MB feature tensor: non-temporal load (TH=NT)
            // keeps WGP$/L2 free for W1 / x reuse.
            v2f a = __builtin_nontemporal_load((const v2f*)(arow + k));
            const int kb = k + khalf;
            v2f bl, bh;
            bl.x = sW1[kb * HID + n0];
            bl.y = sW1[kb * HID + HID + n0];
            bh.x = sW1[kb * HID + 16 + n0];
            bh.y = sW1[kb * HID + HID + 16 + n0];
            acc0 = __builtin_amdgcn_wmma_f32_16x16x4_f32(
                false, a, false, bl, (short)0, acc0, false, false);
            acc1 = __builtin_amdgcn_wmma_f32_16x16x4_f32(
                false, a, false, bh, (short)0, acc1, false, false);
        }

        // C/D layout: VGPR i -> row (i + 8*(lane>=16)), col lane&15.
        // Add bias, ReLU, stage h tile (16x32) into this wave's LDS slice.
        float* shw = sH + wave * (16 * HID);
        #pragma unroll
        for (int i = 0; i < 8; ++i) {
            const int mm = i + ((lane >> 4) << 3);
            float h0 = acc0[i] + sb1[n0];
            float h1 = acc1[i] + sb1[16 + n0];
            shw[mm * HID + n0]      = h0 > 0.0f ? h0 : 0.0f;
            shw[mm * HID + 16 + n0] = h1 > 0.0f ? h1 : 0.0f;
        }
        // Same-wave LDS RAW: DS ops are in-order per wave, but be explicit.
        asm volatile("s_wait_dscnt 0x0" ::: "memory");

        // fc2: x[m][o] = b2[o] + sum_n h[m][n]*W2[n][o]
        // lane -> row (lane&15), output half (lane>>4)*8 .. +7
        const int mm   = lane & 15;
        const int half = lane >> 4;
        const float* hrow = shw + mm * HID;
        float* xrow = x + (size_t)(row0 + mm) * OUT_DIM;
        #pragma unroll
        for (int oo = 0; oo < 8; ++oo) {
            const int o = half * 8 + oo;
            float s = sb2[o];
            #pragma unroll
            for (int nn = 0; nn < HID; ++nn)
                s += hrow[nn] * sW2[nn * OUT_DIM + o];
            xrow[o] = s;
        }
    }
}

// SpMM scatter-add: out[r] += w_e * x[c_e]. 16 lanes per edge ->
// coalesced 64B row gathers (x is L2-resident, regular-temporal) +
// f32 atomics into L2-resident out. Edge lists are stream-once -> NT.
__global__ __launch_bounds__(256)
void spmm_scatter_kernel(const int* __restrict__ edge_row,
                         const int* __restrict__ edge_col,
                         const float* __restrict__ edge_w,
                         const float* __restrict__ x,
                         float* __restrict__ out, int E)
{
    const long t = (long)blockIdx.x * 256 + threadIdx.x;
    const long e = t >> 4;
    const int  o = (int)(t & 15);
    if (e < E) {
        const int   r = __builtin_nontemporal_load(edge_row + e);
        const int   c = __builtin_nontemporal_load(edge_col + e);
        const float w = __builtin_nontemporal_load(edge_w + e);
        atomicAdd(out + (size_t)r * OUT_DIM + o, w * x[(size_t)c * OUT_DIM + o]);
    }
}

extern "C" void kernel_launch(void* const* d_in, const int* in_sizes, int n_in,
                              void* d_out, int out_size, void* d_ws, size_t ws_size,
                              hipStream_t stream) {
    const float* feat = (const float*)d_in[0];
    const int*   er   = (const int*)  d_in[1];
    const int*   ec   = (const int*)  d_in[2];
    const float* ew   = (const float*)d_in[3];
    const float* W1   = (const float*)d_in[4];
    const float* b1   = (const float*)d_in[5];
    const float* W2   = (const float*)d_in[6];
    const float* b2   = (const float*)d_in[7];
    float* out = (float*)d_out;
    float* x   = (float*)d_ws;          // N*16 fp32 intermediate (6.4 MB)

    const int n = in_sizes[0] / IN_DIM; // 100000
    const int E = in_sizes[1];          // 3200000

    // Zero accumulation target (graph-capture-safe memset node).
    hipMemsetAsync(d_out, 0, (size_t)out_size * sizeof(float), stream);

    const int mlp_blocks = (n + 127) / 128;
    mlp_fused_kernel<<<mlp_blocks, 256, SMEM_BYTES, stream>>>(
        feat, W1, b1, W2, b2, x, n);

    const long sthreads = (long)E * 16;
    const int  sblocks  = (int)((sthreads + 255) / 256);
    spmm_scatter_kernel<<<sblocks, 256, 0, stream>>>(er, ec, ew, x, out, E);
}